// STFModelTorch_68015102100077
// MI455X (gfx1250) — compile-verified
//
#include <hip/hip_runtime.h>
#include <hip/hip_bf16.h>
#include <stdint.h>

#define N_DATA   128
#define SEG_LEN  131072
#define N_MAT    8
#define N_MATP   8
#define N_PROCP  5
#define RGASF    8.314f
#define BLOCK    1024
#define TILE     4096                 // 4 elements per thread per tile
#define NTILES   (SEG_LEN / TILE)     // 32

typedef __attribute__((ext_vector_type(2))) float v2f;
typedef __attribute__((ext_vector_type(8))) float v8f;

typedef __attribute__((address_space(1))) int* gint_p;
typedef __attribute__((address_space(3))) int* lint_p;

// ---------- gfx1250 async global->LDS copy (ASYNCcnt-tracked) ----------
__device__ __forceinline__ void async_copy_b32(void* lds_dst, const void* gsrc) {
#if __has_builtin(__builtin_amdgcn_global_load_async_to_lds_b32)
  __builtin_amdgcn_global_load_async_to_lds_b32(
      (gint_p)(uintptr_t)gsrc,       // addrspace(1): flat==global numerically
      (lint_p)lds_dst,               // addrspacecast generic -> LDS
      0, 0);
#else
  unsigned lds_off = (unsigned)(uintptr_t)(lint_p)lds_dst;
  asm volatile("global_load_async_to_lds_b32 %0, %1, off"
               :: "v"(lds_off), "v"(gsrc) : "memory");
#endif
}

__device__ __forceinline__ void wait_async0() {
#if __has_builtin(__builtin_amdgcn_s_wait_asynccnt)
  __builtin_amdgcn_s_wait_asynccnt(0);
#else
  asm volatile("s_wait_asynccnt 0" ::: "memory");
#endif
}

// cp[] layout: 0 K0, 1 alpha1, 2 L0, 3 G200, 4 BetaD*D, 5 SigmaC-Mfda*P,
//              6 Sigma0, 7 R, 8 1/(B0*l0+eps), 9 A0, 10 (xmax-xmin), 11 xmin,
//              12 ymin, 13 1/(ymax-ymin)
__global__ __launch_bounds__(BLOCK) void stf_row_scan_kernel(
    const float* __restrict__ x, const float* __restrict__ pc,
    const float* __restrict__ raw, const float* __restrict__ lb,
    const float* __restrict__ ub, const float* __restrict__ scal,
    const int* __restrict__ matidx, float* __restrict__ out)
{
  __shared__ float4 sbuf4[2][TILE / 4];   // double-buffered x tile (2 x 16 KB)
  __shared__ float  s_tot[32];            // per-wave tile totals
  __shared__ float  s_scan[32];           // WMMA scan result (col0 | col1)
  __shared__ float  cp[16];               // per-row constants

  const int row  = blockIdx.x;
  const int tid  = threadIdx.x;
  const int lane = tid & 31;
  const int wid  = tid >> 5;
  const float* __restrict__ xrow = x   + (size_t)row * SEG_LEN;
  float* __restrict__       orow = out + (size_t)row * SEG_LEN;
  float* sfl0 = (float*)&sbuf4[0][0];
  float* sfl1 = (float*)&sbuf4[1][0];

  if (tid == 0) {
    auto xv = [&](int i) {
      float s = 1.0f / (1.0f + __expf(-raw[i]));          // sigmoid
      return s * (ub[i] - lb[i]) + lb[i];
    };
    const int pb = N_MAT * N_MATP + row * N_PROCP;
    const float SigmaC = xv(pb + 0), K0 = xv(pb + 1), alpha1 = xv(pb + 2);
    const float L0 = xv(pb + 3), G200 = xv(pb + 4);
    const int mb = matidx[row] * N_MATP;
    const float Sigma0 = xv(mb + 0), BetaD = xv(mb + 1), Ea = xv(mb + 2);
    const float Mfda = xv(mb + 3), Di = xv(mb + 4), A0 = xv(mb + 5);
    const float B0 = xv(mb + 6), l0 = xv(mb + 7);
    const float R = pc[row * 4 + 0], Tk = pc[row * 4 + 1], P = pc[row * 4 + 2];
    const float D = Di * __expf(-Ea / (RGASF * Tk));
    const float xmin = scal[0], xmax = scal[1], ymin = scal[2], ymax = scal[3];
    cp[0] = K0;  cp[1] = alpha1; cp[2] = L0;     cp[3] = G200;
    cp[4] = BetaD * D;           cp[5] = SigmaC - Mfda * P;
    cp[6] = Sigma0;              cp[7] = R;
    cp[8] = __builtin_amdgcn_rcpf(B0 * l0 + 1e-9f);
    cp[9] = A0;
    cp[10] = xmax - xmin;        cp[11] = xmin;
    cp[12] = ymin;               cp[13] = __builtin_amdgcn_rcpf(ymax - ymin);
    orow[0] = (K0 - ymin) * cp[13];                       // fit[0] = K0
  }

  // Prologue: async-stage tile 0 (lane-coalesced, 4 dwords per thread)
  #pragma unroll
  for (int k = 0; k < 4; ++k)
    async_copy_b32(&sfl0[k * BLOCK + tid], &xrow[k * BLOCK + tid]);
  wait_async0();
  __syncthreads();

  // Hoist per-row constants into registers (LDS -> VGPR once)
  const float rK0 = cp[0], rAl = cp[1], rL0 = cp[2], rG2 = cp[3];
  const float rBD = cp[4], rSC = cp[5], rS0 = cp[6], rR  = cp[7];
  const float rIB = cp[8], rA0 = cp[9], rXa = cp[10], rXb = cp[11];
  const float rYm = cp[12], rYs = cp[13];

  // Lane-constant WMMA operands: A = L16 (lower-triangular ones), K-split x4.
  const int   m    = lane & 15, h = lane >> 4;
  const float selB = (m < 2) ? 1.0f : 0.0f;       // only cols 0,1 of B non-zero
  const int   bb   = 16 * (m & 1) + 2 * h;        // s_tot base index, <= 18
  const v2f A0m = { (0  + 2*h     <= m) ? 1.0f : 0.0f, (0  + 2*h + 1 <= m) ? 1.0f : 0.0f };
  const v2f A1m = { (4  + 2*h     <= m) ? 1.0f : 0.0f, (4  + 2*h + 1 <= m) ? 1.0f : 0.0f };
  const v2f A2m = { (8  + 2*h     <= m) ? 1.0f : 0.0f, (8  + 2*h + 1 <= m) ? 1.0f : 0.0f };
  const v2f A3m = { (12 + 2*h     <= m) ? 1.0f : 0.0f, (12 + 2*h + 1 <= m) ? 1.0f : 0.0f };

  float carry = 0.0f;
  for (int Tt = 0; Tt < NTILES; ++Tt) {
    const int cur  = Tt & 1;
    const int base = Tt * TILE;
    float* snx = cur ? sfl0 : sfl1;
    if (Tt + 1 < NTILES) {                        // prefetch next tile (async DMA)
      #pragma unroll
      for (int k = 0; k < 4; ++k)
        async_copy_b32(&snx[k * BLOCK + tid], &xrow[base + TILE + k * BLOCK + tid]);
    }

    // 4 consecutive elements per thread, aligned b128 LDS read
    const float4 xq = sbuf4[cur][tid];
    float xnext;
    if (tid < BLOCK - 1) xnext = sbuf4[cur][tid + 1].x;
    else                 xnext = (base + TILE < SEG_LEN) ? xrow[base + TILE] : 0.0f;

    const float xs[5] = { xq.x, xq.y, xq.z, xq.w, xnext };
    float p[4];
    float run = 0.0f;
    #pragma unroll
    for (int k = 0; k < 4; ++k) {
      const int j = base + 4 * tid + k;
      float v = 0.0f;
      if (j < SEG_LEN - 1) {
        const float tp = xs[k]     * rXa + rXb;
        const float tn = xs[k + 1] * rXa + rXb;
        const float u  = tp * (1.0f / 200.0f) + 0.001f;
        const float Lg = rG2 * __powf(u, rAl) + rL0;
        const float beta = rBD * __builtin_amdgcn_rcpf(rR * Lg + 1e-9f);
        const float sig  = (rSC + rS0 * beta * __builtin_amdgcn_rcpf(Lg)) *
                           __builtin_amdgcn_rcpf(1.0f + beta);
        const float s = sig + rA0 * __expf(-tp * rIB);
        v = s * (tn - tp);                        // s * dT
      }
      run += v;
      p[k] = run;                                 // thread-local inclusive prefix
    }

    // ---- intra-wave inclusive scan of thread sums (wave32, 5 shuffles) ----
    float vi = run;
    #pragma unroll
    for (int d = 1; d < 32; d <<= 1) {
      float w = __shfl_up(vi, d, 32);
      if (lane >= d) vi += w;
    }
    const float lane_excl = vi - run;
    if (lane == 31) s_tot[wid] = vi;
    __syncthreads();

    // ---- cross-wave scan of 32 totals via WMMA: D = L16 x [w_lo | w_hi] ----
    if (__builtin_amdgcn_readfirstlane(wid) == 0) {       // wave 0, EXEC all-1s
      v8f acc = {0.f, 0.f, 0.f, 0.f, 0.f, 0.f, 0.f, 0.f};
      v2f B0m = { s_tot[bb + 0]  * selB, s_tot[bb + 1]  * selB };
      acc = __builtin_amdgcn_wmma_f32_16x16x4_f32(false, A0m, false, B0m, (short)0, acc, false, false);
      v2f B1m = { s_tot[bb + 4]  * selB, s_tot[bb + 5]  * selB };
      acc = __builtin_amdgcn_wmma_f32_16x16x4_f32(false, A1m, false, B1m, (short)0, acc, false, false);
      v2f B2m = { s_tot[bb + 8]  * selB, s_tot[bb + 9]  * selB };
      acc = __builtin_amdgcn_wmma_f32_16x16x4_f32(false, A2m, false, B2m, (short)0, acc, false, false);
      v2f B3m = { s_tot[bb + 12] * selB, s_tot[bb + 13] * selB };
      acc = __builtin_amdgcn_wmma_f32_16x16x4_f32(false, A3m, false, B3m, (short)0, acc, false, false);
      if (m < 2) {                                        // D col m: rows 8h..8h+7
        #pragma unroll
        for (int r = 0; r < 8; ++r) s_scan[16 * m + 8 * h + r] = acc[r];
      }
    }
    __syncthreads();

    // scan32(i) = s_scan[i] + (i>=16 ? scan_lo_total : 0)
    float wave_excl = 0.0f;
    if (wid > 0) {
      const int i = wid - 1;
      wave_excl = s_scan[i] + ((i >= 16) ? s_scan[15] : 0.0f);
    }
    const float tilesum = s_scan[31] + s_scan[15];

    const float ybase = rK0 + carry + wave_excl + lane_excl;
    #pragma unroll
    for (int k = 0; k < 4; ++k) {
      const int j = base + 4 * tid + k;
      if (j < SEG_LEN - 1)
        orow[j + 1] = (ybase + p[k] - rYm) * rYs;         // K0 + cumsum, rescaled
    }
    carry += tilesum;

    wait_async0();                                        // next tile landed
    __syncthreads();                                      // LDS reuse fence
  }
}

extern "C" void kernel_launch(void* const* d_in, const int* in_sizes, int n_in,
                              void* d_out, int out_size, void* d_ws, size_t ws_size,
                              hipStream_t stream) {
  (void)in_sizes; (void)n_in; (void)out_size; (void)d_ws; (void)ws_size;
  const float* x    = (const float*)d_in[0];   // x_scaled
  const float* pc   = (const float*)d_in[1];   // process_c
  const float* raw  = (const float*)d_in[2];   // raw_params
  const float* lb   = (const float*)d_in[3];   // para_lb
  const float* ub   = (const float*)d_in[4];   // para_ub
  const float* scal = (const float*)d_in[5];   // scaler_params
  // d_in[6] = fit_index (unused: layout is implicit row-major)
  const int*   mat  = (const int*)d_in[7];     // material_index
  float* out = (float*)d_out;
  hipLaunchKernelGGL(stf_row_scan_kernel, dim3(N_DATA), dim3(BLOCK), 0, stream,
                     x, pc, raw, lb, ub, scal, mat, out);
}